// Reg_3stage_6064493822532
// MI455X (gfx1250) — compile-verified
//
#include <hip/hip_runtime.h>
#include <hip/hip_bf16.h>

// ---------------------------------------------------------------------------
// Reg_3stage / Classifier3Stage for MI455X (gfx1250, wave32).
//  K0: [128,N] -> [N,128] transpose (coalesced LDS tile transpose)
//  K1: stage1 dense per-row MLP via V_WMMA_F32_16X16X4_F32 + lane argmax
//  K2: stage2 CondMul chain, 1 wave per pixel (readlane bcast, bpermute argmax)
//  K3: stage3 CondMul chain, same structure, writes final class index
// ---------------------------------------------------------------------------

#define HH   112
#define WW   320
#define NPIX 35840            // HH*WW (== B*H*W, B=1)
#define LSTR 36               // padded LDS row stride (floats), keeps 8B align

typedef __attribute__((ext_vector_type(2))) float v2f;
typedef __attribute__((ext_vector_type(8))) float v8f;

__device__ __forceinline__ v8f wmma4(v2f a, v2f b, v8f c) {
  // D = A(16x4 f32) * B(4x16 f32) + C(16x16 f32)
  return __builtin_amdgcn_wmma_f32_16x16x4_f32(false, a, false, b, (short)0, c,
                                               false, false);
}

__device__ __forceinline__ float lrelu(float v) {
  return v >= 0.0f ? v : 0.01f * v;
}

__device__ __forceinline__ float readlane_f(float v, int l) {
  return __builtin_bit_cast(float,
      __builtin_amdgcn_readlane(__builtin_bit_cast(int, v), l));
}

__device__ __forceinline__ float bperm_f(float v, int srcLane) {
  return __builtin_bit_cast(float,
      __builtin_amdgcn_ds_bpermute(srcLane << 2, __builtin_bit_cast(int, v)));
}

__device__ __forceinline__ int bperm_i(int v, int srcLane) {
  return __builtin_amdgcn_ds_bpermute(srcLane << 2, v);
}

// --------------------------- K0: transpose --------------------------------
// x   : [128, NPIX]  (original [1,128,112,320] flattened)
// xT  : [NPIX, 128]
__global__ __launch_bounds__(256) void k_transpose(const float* __restrict__ x,
                                                   float* __restrict__ xT) {
  __shared__ float tile[32][33];
  const int p0 = blockIdx.x * 32;
  const int c0 = blockIdx.y * 32;
#pragma unroll
  for (int j = 0; j < 4; ++j) {
    const int c = c0 + threadIdx.y + j * 8;
    tile[threadIdx.y + j * 8][threadIdx.x] =
        x[(size_t)c * NPIX + p0 + threadIdx.x];
  }
  __syncthreads();
#pragma unroll
  for (int j = 0; j < 4; ++j) {
    const int p = p0 + threadIdx.y + j * 8;
    xT[(size_t)p * 128 + c0 + threadIdx.x] = tile[threadIdx.x][threadIdx.y + j * 8];
  }
}

// --------------------------- K1: stage 1 (WMMA) ----------------------------
// Per block: one image row h (320 px). 4 waves x 5 M-tiles of 16 px.
// Layer0: [16x128]x[128->32], Layer1: [16x32]x[32->32], Layer2: [16x32]x[32->16]
// w0:[H,32,128] b0:[H,32]  w1:[H,32,32] b1:[H,32]  w2:[H,16,32] b2:[H,16]
__global__ __launch_bounds__(128) void k_stage1(
    const float* __restrict__ x, const float* __restrict__ w0,
    const float* __restrict__ b0, const float* __restrict__ w1,
    const float* __restrict__ b1, const float* __restrict__ w2,
    const float* __restrict__ b2, int* __restrict__ inds1) {
  __shared__ float lbuf[4][16 * LSTR];
  const int h    = blockIdx.x;
  const int tid  = threadIdx.x;
  const int wv   = tid >> 5;
  const int lane = tid & 31;
  const int half = lane >> 4;   // 0: lanes 0-15, 1: lanes 16-31
  const int lm   = lane & 15;
  float* buf = lbuf[wv];

  const float* w0h = w0 + (size_t)h * 32 * 128;
  const float* w1h = w1 + (size_t)h * 32 * 32;
  const float* w2h = w2 + (size_t)h * 16 * 32;

  for (int t = wv; t < 20; t += 4) {
    const int wpix = t * 16;
    const size_t pbase = (size_t)h * WW + wpix + lm;

    // ---------------- layer 0: K=128, out 32 (two 16-wide halves) ----------
    float bn0 = b0[h * 32 + lm];
    float bn1 = b0[h * 32 + 16 + lm];
    v8f acc0, acc1;
#pragma unroll
    for (int r = 0; r < 8; ++r) { acc0[r] = bn0; acc1[r] = bn1; }

#pragma unroll
    for (int k = 0; k < 128; k += 4) {
      const int c0 = k + 2 * half;
      v2f a;  // A[16x4]: lane<16 -> (m=lm, k+0/k+1), lane>=16 -> (m=lm, k+2/k+3)
      a.x = x[(size_t)c0 * NPIX + pbase];
      a.y = x[(size_t)(c0 + 1) * NPIX + pbase];
      const float2 bv0 = *reinterpret_cast<const float2*>(w0h + (size_t)lm * 128 + c0);
      const float2 bv1 = *reinterpret_cast<const float2*>(w0h + (size_t)(16 + lm) * 128 + c0);
      v2f b0v; b0v.x = bv0.x; b0v.y = bv0.y;
      v2f b1v; b1v.x = bv1.x; b1v.y = bv1.y;
      acc0 = wmma4(a, b0v, acc0);
      acc1 = wmma4(a, b1v, acc1);
    }

    // lrelu + deposit T0[16x32] to LDS (C layout -> row-major tile)
#pragma unroll
    for (int r = 0; r < 8; ++r) {
      const int m = r + 8 * half;
      buf[m * LSTR + lm]      = lrelu(acc0[r]);
      buf[m * LSTR + 16 + lm] = lrelu(acc1[r]);
    }
    __asm__ volatile("s_wait_dscnt 0" ::: "memory");

    // ---------------- layer 1: K=32, out 32 --------------------------------
    bn0 = b1[h * 32 + lm];
    bn1 = b1[h * 32 + 16 + lm];
    v8f t10, t11;
#pragma unroll
    for (int r = 0; r < 8; ++r) { t10[r] = bn0; t11[r] = bn1; }
#pragma unroll
    for (int k = 0; k < 32; k += 4) {
      const int c0 = k + 2 * half;
      const float2 av = *reinterpret_cast<const float2*>(&buf[lm * LSTR + c0]);
      v2f a; a.x = av.x; a.y = av.y;
      const float2 bv0 = *reinterpret_cast<const float2*>(w1h + lm * 32 + c0);
      const float2 bv1 = *reinterpret_cast<const float2*>(w1h + (16 + lm) * 32 + c0);
      v2f b0v; b0v.x = bv0.x; b0v.y = bv0.y;
      v2f b1v; b1v.x = bv1.x; b1v.y = bv1.y;
      t10 = wmma4(a, b0v, t10);
      t11 = wmma4(a, b1v, t11);
    }
    __asm__ volatile("s_wait_dscnt 0" ::: "memory");

    // lrelu + deposit T1[16x32]
#pragma unroll
    for (int r = 0; r < 8; ++r) {
      const int m = r + 8 * half;
      buf[m * LSTR + lm]      = lrelu(t10[r]);
      buf[m * LSTR + 16 + lm] = lrelu(t11[r]);
    }
    __asm__ volatile("s_wait_dscnt 0" ::: "memory");

    // ---------------- layer 2: K=32, out 16 (no activation) ----------------
    const float bn2 = b2[h * 16 + lm];
    v8f t2;
#pragma unroll
    for (int r = 0; r < 8; ++r) t2[r] = bn2;
#pragma unroll
    for (int k = 0; k < 32; k += 4) {
      const int c0 = k + 2 * half;
      const float2 av = *reinterpret_cast<const float2*>(&buf[lm * LSTR + c0]);
      v2f a; a.x = av.x; a.y = av.y;
      const float2 bv = *reinterpret_cast<const float2*>(w2h + lm * 32 + c0);
      v2f bvv; bvv.x = bv.x; bvv.y = bv.y;
      t2 = wmma4(a, bvv, t2);
    }

    // argmax over 16 classes (lanes within each 16-half), first-max tiebreak
#pragma unroll
    for (int r = 0; r < 8; ++r) {
      float v = t2[r];
      int idx = lm;
#pragma unroll
      for (int mask = 1; mask <= 8; mask <<= 1) {
        const float ov = bperm_f(v, lane ^ mask);
        const int   oi = bperm_i(idx, lane ^ mask);
        if (ov > v || (ov == v && oi < idx)) { v = ov; idx = oi; }
      }
      if (lm == 0) {
        const int m = r + 8 * half;
        inds1[h * WW + wpix + m] = idx;
      }
    }
    __asm__ volatile("s_wait_dscnt 0" ::: "memory");
  }
}

// --------------------------- K2/K3: CondMul chain --------------------------
// One wave per pixel. lane = out channel. wA:[K,128,32] wB,wC:[K,32,32]
// stage==2: route = inds1 + 16*h ; writes raw inds12 (unclipped)
// stage==3: route = clip(inds12,0,255) + 256*h ; writes final float index
__global__ __launch_bounds__(256) void k_condmul(
    const float* __restrict__ xT, const float* __restrict__ wA,
    const float* __restrict__ bA, const float* __restrict__ wB,
    const float* __restrict__ bB, const float* __restrict__ wC,
    const float* __restrict__ bC, const int* __restrict__ route_in,
    int* __restrict__ route_out, float* __restrict__ out, int stage) {
  const int lane = threadIdx.x & 31;
  const int wv   = threadIdx.x >> 5;
  const int pix  = blockIdx.x * 8 + wv;
  if (pix >= NPIX) return;
  const int h = pix / WW;

  const int base_idx = route_in[pix];
  int route;
  if (stage == 2) {
    route = base_idx + 16 * h;                       // K = 1792
  } else {
    int c = base_idx < 0 ? 0 : (base_idx > 255 ? 255 : base_idx);
    route = c + 256 * h;                             // K = 28672
  }

  // coalesced activation row load
  const float* xr = xT + (size_t)pix * 128;
  const float xv0 = xr[lane];
  const float xv1 = xr[32 + lane];
  const float xv2 = xr[64 + lane];
  const float xv3 = xr[96 + lane];

  // layer A: 128 -> 32
  const float* wa = wA + (size_t)route * (128 * 32);
  float acc = bA[(size_t)route * 32 + lane];
#pragma unroll
  for (int i = 0; i < 128; ++i) {
    float xi;
    if (i < 32)       xi = readlane_f(xv0, i);
    else if (i < 64)  xi = readlane_f(xv1, i - 32);
    else if (i < 96)  xi = readlane_f(xv2, i - 64);
    else              xi = readlane_f(xv3, i - 96);
    acc = fmaf(xi, wa[(size_t)i * 32 + lane], acc);
  }
  acc = lrelu(acc);

  // layer B: 32 -> 32
  const float* wb = wB + (size_t)route * 1024;
  float acc2 = bB[(size_t)route * 32 + lane];
#pragma unroll
  for (int i = 0; i < 32; ++i)
    acc2 = fmaf(readlane_f(acc, i), wb[i * 32 + lane], acc2);
  acc2 = lrelu(acc2);

  // layer C: 32 -> 32 (no activation)
  const float* wc = wC + (size_t)route * 1024;
  float acc3 = bC[(size_t)route * 32 + lane];
#pragma unroll
  for (int i = 0; i < 32; ++i)
    acc3 = fmaf(readlane_f(acc2, i), wc[i * 32 + lane], acc3);

  // argmax over 32 lanes, first-max tiebreak
  float v = acc3;
  int idx = lane;
#pragma unroll
  for (int mask = 16; mask >= 1; mask >>= 1) {
    const float ov = bperm_f(v, lane ^ mask);
    const int   oi = bperm_i(idx, lane ^ mask);
    if (ov > v || (ov == v && oi < idx)) { v = ov; idx = oi; }
  }

  if (lane == 0) {
    if (stage == 2) {
      route_out[pix] = base_idx * 16 + idx - 8;      // raw inds12 (unclipped)
    } else {
      int v123 = base_idx * 16 + idx - 8;            // uses UNCLIPPED inds12
      v123 = v123 < 0 ? 0 : (v123 > 4095 ? 4095 : v123);
      out[pix] = (float)v123;
    }
  }
}

// ---------------------------------------------------------------------------
extern "C" void kernel_launch(void* const* d_in, const int* in_sizes, int n_in,
                              void* d_out, int out_size, void* d_ws,
                              size_t ws_size, hipStream_t stream) {
  (void)in_sizes; (void)n_in; (void)out_size; (void)ws_size;
  const float* x    = (const float*)d_in[0];
  const float* w1_0 = (const float*)d_in[1];
  const float* b1_0 = (const float*)d_in[2];
  const float* w1_1 = (const float*)d_in[3];
  const float* b1_1 = (const float*)d_in[4];
  const float* w1_2 = (const float*)d_in[5];
  const float* b1_2 = (const float*)d_in[6];
  const float* w2_0 = (const float*)d_in[7];
  const float* b2_0 = (const float*)d_in[8];
  const float* w2_1 = (const float*)d_in[9];
  const float* b2_1 = (const float*)d_in[10];
  const float* w2_2 = (const float*)d_in[11];
  const float* b2_2 = (const float*)d_in[12];
  const float* w3_0 = (const float*)d_in[13];
  const float* b3_0 = (const float*)d_in[14];
  const float* w3_1 = (const float*)d_in[15];
  const float* b3_1 = (const float*)d_in[16];
  const float* w3_2 = (const float*)d_in[17];
  const float* b3_2 = (const float*)d_in[18];
  float* out = (float*)d_out;

  // workspace layout: xT (18,350,080 B) | inds1 (143,360 B) | inds12 (143,360 B)
  char* ws = (char*)d_ws;
  float* xT    = (float*)ws;
  int*   inds1 = (int*)(ws + (size_t)NPIX * 128 * sizeof(float));
  int*   inds12 = (int*)(ws + (size_t)NPIX * 128 * sizeof(float) +
                         (size_t)NPIX * sizeof(int));

  k_transpose<<<dim3(NPIX / 32, 4), dim3(32, 8), 0, stream>>>(x, xT);
  k_stage1<<<HH, 128, 0, stream>>>(x, w1_0, b1_0, w1_1, b1_1, w1_2, b1_2, inds1);
  k_condmul<<<NPIX / 8, 256, 0, stream>>>(xT, w2_0, b2_0, w2_1, b2_1, w2_2, b2_2,
                                          inds1, inds12, nullptr, 2);
  k_condmul<<<NPIX / 8, 256, 0, stream>>>(xT, w3_0, b3_0, w3_1, b3_1, w3_2, b3_2,
                                          inds12, nullptr, out, 3);
}